// ManipulationModel_16363825398055
// MI455X (gfx1250) — compile-verified
//
#include <hip/hip_runtime.h>
#include <stdint.h>

// ---- problem constants (from reference) ----
#define B_     8
#define H_     480
#define W_     640
#define N_     (H_ * W_)          // 307200
#define RWIN   15
#define WIN    31                 // 2R+1
#define PATCH  (WIN * WIN)        // 961
#define SORTN  1024
#define BLK    256
#define VPT    4                  // pixels per thread (one async b128 each)
#define PBLK   (BLK * VPT)        // 1024 pixels per block
#define NBLK2  (N_ / PBLK)        // 300 blocks per batch
#define NPTS   8192
#define HALF   0.15f
#define NORMS  (2.0f / 0.3f)
#define INV_F  (1.0f / 615.0f)
#define INV_SC (1.0f / 1000.0f)

typedef uint32_t u32x4 __attribute__((ext_vector_type(4)));
typedef uint32_t u32x8 __attribute__((ext_vector_type(8)));

// ---------------------------------------------------------------------------
// back-projection + crop-box mask
// ---------------------------------------------------------------------------
__device__ __forceinline__ int pix_mask(int n, float d, const float* gi,
                                        float& x, float& y, float& z) {
    int v = n / W_;
    int u = n - v * W_;
    z = d * INV_SC;
    x = ((float)u - 320.0f) * INV_F * z;
    y = ((float)v - 240.0f) * INV_F * z;
    return (fabsf(x - gi[0]) <= HALF) &&
           (fabsf(y - gi[1]) <= HALF) &&
           (fabsf(z - gi[2]) <= HALF);
}

// ---------------------------------------------------------------------------
// kernel 0: zero-fill output (padding slots must be exactly 0)
// ---------------------------------------------------------------------------
__global__ void zero_kernel(float* __restrict__ out, int total) {
    int i = blockIdx.x * blockDim.x + threadIdx.x;
    if (i < total) out[i] = 0.0f;
}

// ---------------------------------------------------------------------------
// kernel 1: gaze depth median-fill + gaze3d.
// 31x31 depth tile is DMA'd to LDS via the Tensor Data Mover (TDM), then a
// 1024-wide LDS bitonic sort gives the lower-median of positives.
// ginfo[b] = {gaze3d.x, gaze3d.y, gaze3d.z, filled_depth}
// ---------------------------------------------------------------------------
__global__ void fill_gaze_kernel(const float* __restrict__ depth,
                                 const int*   __restrict__ gaze,
                                 float*       __restrict__ ginfo) {
    int b   = blockIdx.x;
    int tid = threadIdx.x;

    __shared__ float patch[PATCH];
    __shared__ float key[SORTN];
    __shared__ int   scount;

    int gx = gaze[b * 2 + 0];
    int gy = gaze[b * 2 + 1];
    // dynamic_slice-style clamped window start
    int sx = gx - RWIN; sx = sx < 0 ? 0 : sx; sx = sx > (W_ - WIN) ? (W_ - WIN) : sx;
    int sy = gy - RWIN; sy = sy < 0 ? 0 : sy; sy = sy > (H_ - WIN) ? (H_ - WIN) : sy;

    if (tid < 32) {  // wave 0 only issues the TDM descriptor
        uint32_t lds = (uint32_t)(uintptr_t)&patch[0];
        uint64_t ga  = (uint64_t)(uintptr_t)(depth + (size_t)b * N_ +
                                             (size_t)sy * W_ + sx);
        // D# group0: count=1 valid, lds_addr, global_addr[56:0], type=2
        u32x4 g0 = { 1u,
                     lds,
                     (uint32_t)ga,
                     (uint32_t)((ga >> 32) & 0x01FFFFFFu) | (2u << 30) };
        // D# group1: data_size=4B; tensor 640x480; tile 31x31; dim0 stride 640
        u32x8 g1 = { 2u << 16,                 // data_size = 2 (4 bytes)
                     (uint32_t)W_ << 16,       // tensor_dim0[15:0] @ bits[63:48]
                     (uint32_t)H_ << 16,       // dim0 hi=0 | tensor_dim1[15:0]
                     (uint32_t)WIN << 16,      // dim1 hi=0 | tile_dim0 = 31
                     (uint32_t)WIN,            // tile_dim1 = 31, tile_dim2 = 0
                     (uint32_t)W_,             // tensor_dim0_stride = 640
                     0u, 0u };
        u32x4 g2 = { 0u, 0u, 0u, 0u };         // 2D tile: groups 2/3 unused
        u32x4 g3 = { 0u, 0u, 0u, 0u };
        asm volatile("tensor_load_to_lds %0, %1, %2, %3"
                     :: "s"(g0), "s"(g1), "s"(g2), "s"(g3) : "memory");
        __builtin_amdgcn_s_wait_tensorcnt(0);
    }
    __syncthreads();   // tile now visible to all 8 waves

    // build sort keys: positives keep value, non-positives / pad -> +inf
    for (int i = tid; i < SORTN; i += BLK) {
        float v = (i < PATCH) ? patch[i] : 0.0f;
        key[i] = (v > 0.0f) ? v : __builtin_inff();
    }
    if (tid == 0) scount = 0;
    __syncthreads();

    int c = 0;
    for (int i = tid; i < PATCH; i += BLK)
        if (patch[i] > 0.0f) c++;
    atomicAdd(&scount, c);

    // bitonic sort ascending (1024 elements, 256 threads)
    for (int k = 2; k <= SORTN; k <<= 1) {
        for (int j = k >> 1; j > 0; j >>= 1) {
            __syncthreads();
            for (int t = tid; t < SORTN; t += BLK) {
                int ixj = t ^ j;
                if (ixj > t) {
                    float a = key[t], bb = key[ixj];
                    bool up = ((t & k) == 0);
                    if (up ? (a > bb) : (a < bb)) { key[t] = bb; key[ixj] = a; }
                }
            }
        }
    }
    __syncthreads();

    if (tid == 0) {
        int   n    = scount;
        int   mi   = ((n > 0) ? (n - 1) : 0) >> 1;     // lower median index
        float med  = key[mi];
        float cur  = patch[(gy - sy) * WIN + (gx - sx)];
        float fill = (cur > 0.0f) ? cur : med;
        float z    = fill * INV_SC;
        ginfo[b * 4 + 0] = ((float)gx - 320.0f) * INV_F * z;
        ginfo[b * 4 + 1] = ((float)gy - 240.0f) * INV_F * z;
        ginfo[b * 4 + 2] = z;
        ginfo[b * 4 + 3] = fill;
    }
}

// ---------------------------------------------------------------------------
// kernel 2: per-block in-box counts. Depth plane staged through LDS with a
// single async global->LDS B128 per lane (16B/lane, 1024 px per block).
// ---------------------------------------------------------------------------
__global__ void count_kernel(const float* __restrict__ depth,
                             const int*   __restrict__ gaze,
                             const float* __restrict__ ginfo,
                             int*         __restrict__ counts) {
    int b   = blockIdx.y;
    int blk = blockIdx.x;
    int tid = threadIdx.x;
    int n0  = blk * PBLK + tid * VPT;

    __shared__ float dbuf[PBLK];
    __shared__ int   cnt;
    if (tid == 0) cnt = 0;

    uint32_t la = (uint32_t)(uintptr_t)&dbuf[tid * VPT];
    uint64_t ga = (uint64_t)(uintptr_t)(depth + (size_t)b * N_ + n0);
    asm volatile("global_load_async_to_lds_b128 %0, %1, off"
                 :: "v"(la), "v"(ga) : "memory");
    asm volatile("s_wait_asynccnt 0" ::: "memory");

    int gx = gaze[b * 2 + 0], gy = gaze[b * 2 + 1];
    int gflat = gy * W_ + gx;

    int s = 0;
    #pragma unroll
    for (int k = 0; k < VPT; k++) {
        float d = dbuf[tid * VPT + k];
        int   n = n0 + k;
        if (n == gflat) d = ginfo[b * 4 + 3];   // filled depth at gaze pixel
        float x, y, z;
        s += pix_mask(n, d, &ginfo[b * 4], x, y, z);
    }

    __syncthreads();
    atomicAdd(&cnt, s);
    __syncthreads();
    if (tid == 0) counts[b * NBLK2 + blk] = cnt;
}

// ---------------------------------------------------------------------------
// kernel 3: per-batch exclusive scan of the 300 block counts (stable order)
// ---------------------------------------------------------------------------
__global__ void scan_kernel(const int* __restrict__ counts,
                            int*       __restrict__ offsets) {
    int b   = blockIdx.x;
    int tid = threadIdx.x;
    __shared__ int buf[BLK];
    __shared__ int carry;
    if (tid == 0) carry = 0;
    __syncthreads();

    for (int base = 0; base < NBLK2; base += BLK) {
        int i = base + tid;
        int v = (i < NBLK2) ? counts[b * NBLK2 + i] : 0;
        buf[tid] = v;
        __syncthreads();
        for (int off = 1; off < BLK; off <<= 1) {   // Hillis-Steele inclusive
            int add = (tid >= off) ? buf[tid - off] : 0;
            __syncthreads();
            buf[tid] += add;
            __syncthreads();
        }
        int total = buf[BLK - 1];
        if (i < NBLK2) offsets[b * NBLK2 + i] = carry + buf[tid] - v;  // exclusive
        __syncthreads();
        if (tid == 0) carry += total;
        __syncthreads();
    }
}

// ---------------------------------------------------------------------------
// kernel 4: rank each in-box pixel (block offset + intra-block exclusive scan,
// preserving row-major order) and scatter normalized XYZ|RGB into slot rank.
// RGB lines for masked pixels are prefetched before the scan barriers.
// ---------------------------------------------------------------------------
__global__ void scatter_kernel(const float* __restrict__ depth,
                               const float* __restrict__ image,
                               const int*   __restrict__ gaze,
                               const float* __restrict__ ginfo,
                               const int*   __restrict__ offsets,
                               float*       __restrict__ out) {
    int b   = blockIdx.y;
    int blk = blockIdx.x;
    int tid = threadIdx.x;
    int n0  = blk * PBLK + tid * VPT;

    __shared__ float dbuf[PBLK];
    __shared__ int   sc[BLK];

    uint32_t la = (uint32_t)(uintptr_t)&dbuf[tid * VPT];
    uint64_t ga = (uint64_t)(uintptr_t)(depth + (size_t)b * N_ + n0);
    asm volatile("global_load_async_to_lds_b128 %0, %1, off"
                 :: "v"(la), "v"(ga) : "memory");
    asm volatile("s_wait_asynccnt 0" ::: "memory");

    int gx = gaze[b * 2 + 0], gy = gaze[b * 2 + 1];
    int gflat = gy * W_ + gx;

    int   m[VPT];
    float px[VPT], py[VPT], pz[VPT];
    int   s = 0;
    #pragma unroll
    for (int k = 0; k < VPT; k++) {
        float d = dbuf[tid * VPT + k];
        int   n = n0 + k;
        if (n == gflat) d = ginfo[b * 4 + 3];
        m[k] = pix_mask(n, d, &ginfo[b * 4], px[k], py[k], pz[k]);
        s += m[k];
    }

    // warm the cache for the gated RGB reads while the scan barriers run
    if (s) {
        size_t ib = (size_t)b * 3 * N_ + (size_t)n0;
        __builtin_prefetch(&image[ib], 0, 0);
        __builtin_prefetch(&image[ib + N_], 0, 0);
        __builtin_prefetch(&image[ib + 2 * N_], 0, 0);
    }

    sc[tid] = s;
    __syncthreads();
    for (int off = 1; off < BLK; off <<= 1) {       // stable intra-block scan
        int add = (tid >= off) ? sc[tid - off] : 0;
        __syncthreads();
        sc[tid] += add;
        __syncthreads();
    }

    int rank = offsets[b * NBLK2 + blk] + sc[tid] - s;
    #pragma unroll
    for (int k = 0; k < VPT; k++) {
        if (m[k] && rank < NPTS) {
            int    n  = n0 + k;
            size_t ob = ((size_t)b * NPTS + (size_t)rank) * 6;
            out[ob + 0] = (px[k] - ginfo[b * 4 + 0]) * NORMS;
            out[ob + 1] = (py[k] - ginfo[b * 4 + 1]) * NORMS;
            out[ob + 2] = (pz[k] - ginfo[b * 4 + 2]) * NORMS;
            size_t ib = (size_t)b * 3 * N_ + (size_t)n;
            out[ob + 3] = image[ib];
            out[ob + 4] = image[ib + N_];
            out[ob + 5] = image[ib + 2 * N_];
        }
        rank += m[k];
    }
}

// ---------------------------------------------------------------------------
extern "C" void kernel_launch(void* const* d_in, const int* in_sizes, int n_in,
                              void* d_out, int out_size, void* d_ws, size_t ws_size,
                              hipStream_t stream) {
    // setup_inputs order: image, depth_, state, gaze, last_action
    const float* image = (const float*)d_in[0];
    const float* depth = (const float*)d_in[1];
    const int*   gaze  = (const int*)d_in[3];
    float*       out   = (float*)d_out;

    // workspace: [B*4 f32 ginfo (pad 64 f32)] [B*NBLK2 counts] [B*NBLK2 offsets]
    float* ginfo   = (float*)d_ws;
    int*   counts  = (int*)((char*)d_ws + 64 * sizeof(float));
    int*   offsets = counts + B_ * NBLK2;

    zero_kernel<<<(out_size + BLK - 1) / BLK, BLK, 0, stream>>>(out, out_size);
    fill_gaze_kernel<<<B_, BLK, 0, stream>>>(depth, gaze, ginfo);
    dim3 grid(NBLK2, B_);
    count_kernel<<<grid, BLK, 0, stream>>>(depth, gaze, ginfo, counts);
    scan_kernel<<<B_, BLK, 0, stream>>>(counts, offsets);
    scatter_kernel<<<grid, BLK, 0, stream>>>(depth, image, gaze, ginfo, offsets, out);
}